// VectorQuantizerLinearDiffable_5282809774150
// MI455X (gfx1250) — compile-verified
//
#include <hip/hip_runtime.h>
#include <math.h>

typedef __attribute__((ext_vector_type(2))) float v2f;
typedef __attribute__((ext_vector_type(4))) float v4f;
typedef __attribute__((ext_vector_type(8))) float v8f;

#define D_DIM 128
#define K_DIM 512
#define M_ROWS 64
#define SROW 516            // 64*516*4B LDS tile, +4 floats/row -> banks shift by 4/row
#define SOFTMIN_B 10.0f
#define BETA_C 0.25f

// ---------------------------------------------------------------------------
// Pass 0: code norms ||e||^2, zero histogram + scalar accumulators
// ---------------------------------------------------------------------------
__global__ __launch_bounds__(512)
void vq_init(const float* __restrict__ emb, float* __restrict__ enorm,
             float* __restrict__ hist, float* __restrict__ accums)
{
  const int t = threadIdx.x;               // 512 threads == K
  const float* e = emb + (size_t)t * D_DIM;
  float s = 0.f;
  for (int d = 0; d < D_DIM; d += 4) {
    v4f v = *(const v4f*)(e + d);
    s += v.x * v.x + v.y * v.y + v.z * v.z + v.w * v.w;
  }
  enorm[t] = s;
  hist[t]  = 0.f;
  if (t < 2) accums[t] = 0.f;
}

// ---------------------------------------------------------------------------
// Main kernel: WMMA distance GEMM + row-wise softmin/argmin + outputs
// ---------------------------------------------------------------------------
__global__ __launch_bounds__(128)
void vq_main(const float* __restrict__ lat, const float* __restrict__ emb,
             const float* __restrict__ enorm, float* __restrict__ hist,
             float* __restrict__ accums,
             float* __restrict__ out_q, float* __restrict__ out_ind,
             float* __restrict__ out_soft)
{
  __shared__ float s_s[M_ROWS * SROW];     // s = ||e||^2 - 2 x.e  (xnorm cancels)
  __shared__ float s_xnorm[M_ROWS];
  __shared__ float s_mn[M_ROWS];
  __shared__ int   s_am[M_ROWS];
  __shared__ float s_inv[M_ROWS];
  __shared__ float s_redf[128];
  __shared__ int   s_redi[128];

  const int tid  = threadIdx.x;
  const int wave = tid >> 5;
  const int lane = tid & 31;
  const size_t rowbase = (size_t)blockIdx.x * M_ROWS;

  // -------- pass 1: V_WMMA_F32_16X16X4_F32, seed C with ||e||^2, A = -2x ----
  {
    const int   mloc  = wave * 16 + (lane & 15);
    const int   khalf = lane >> 4;                       // A layout: K pairs
    const float* xrow = lat + (rowbase + mloc) * D_DIM + 2 * khalf;
    v2f a[32];
    float part = 0.f;
#pragma unroll
    for (int s = 0; s < 32; ++s) {
      v2f x = *(const v2f*)(xrow + 4 * s);
      part += x.x * x.x + x.y * x.y;
      a[s] = -2.f * x;
    }
    float other = __shfl_xor(part, 16, 32);              // combine the two K-halves
    if (lane < 16) s_xnorm[wave * 16 + lane] = part + other;

    for (int nt = 0; nt < 32; ++nt) {                    // 32 tiles of 16 codes
      const int    ncol = nt * 16 + (lane & 15);
      const float* erow = emb + (size_t)ncol * D_DIM + 2 * khalf;
      const float  en   = enorm[ncol];
      v8f c = {en, en, en, en, en, en, en, en};
#pragma unroll
      for (int s = 0; s < 32; ++s) {                     // K = 128 in steps of 4
        v2f b = *(const v2f*)(erow + 4 * s);
        c = __builtin_amdgcn_wmma_f32_16x16x4_f32(false, a[s], false, b,
                                                  (short)0, c, false, false);
      }
      const int rbase = wave * 16 + 8 * khalf;           // D layout: M = v + 8*hi
#pragma unroll
      for (int v = 0; v < 8; ++v)
        s_s[(rbase + v) * SROW + ncol] = c[v];
    }
  }
  __syncthreads();

  // -------- pass 2: per-row argmin + softmin(beta=10) -----------------------
  const int r    = tid & 63;
  const int half = tid >> 6;                             // 2 threads per row
  float* srow = s_s + r * SROW + half * 256;

  float mn = 3.402823e38f;
  int   am = 0;
  for (int i = 0; i < 256; ++i) {
    const int c = (i + r) & 255;                         // bank-conflict-free scan
    const float vv = srow[c];
    if (vv < mn) { mn = vv; am = c; }
  }
  s_redf[tid] = mn;
  s_redi[tid] = am + half * 256;
  __syncthreads();
  if (tid < 64) {
    const float m0 = s_redf[tid], m1 = s_redf[tid + 64];
    const int   a0 = s_redi[tid], a1 = s_redi[tid + 64];
    s_mn[tid] = fminf(m0, m1);
    s_am[tid] = (m1 < m0) ? a1 : a0;
  }
  __syncthreads();

  const float rowmn = s_mn[r];
  float sum = 0.f;
  for (int i = 0; i < 256; ++i) {
    const int c = (i + r) & 255;
    const float e = __expf(SOFTMIN_B * (rowmn - srow[c]));
    srow[c] = e;
    sum += e;
  }
  s_redf[tid] = sum;
  __syncthreads();
  if (tid < 64) s_inv[tid] = 1.f / (s_redf[tid] + s_redf[tid + 64]);
  __syncthreads();

  // coalesced non-temporal stream of the 256MB soft-assignment output
  for (int i = tid; i < M_ROWS * (K_DIM / 4); i += 128) {
    const int rr = i >> 7;
    const int c4 = (i & 127) << 2;
    const float inv = s_inv[rr];
    const float* p = s_s + rr * SROW + c4;
    v4f o = {p[0] * inv, p[1] * inv, p[2] * inv, p[3] * inv};
    __builtin_nontemporal_store(o, (v4f*)(out_soft + (rowbase + rr) * K_DIM + c4));
  }

  // quantized rows (hard one-hot forward) + squared-error partial
  float sq = 0.f;
  for (int i = tid; i < M_ROWS * (D_DIM / 4); i += 128) {
    const int rr = i >> 5;
    const int c4 = (i & 31) << 2;
    const int ind = s_am[rr];
    v4f e4 = *(const v4f*)(emb + (size_t)ind * D_DIM + c4);
    v4f x4 = *(const v4f*)(lat + (rowbase + rr) * D_DIM + c4);
    __builtin_nontemporal_store(e4, (v4f*)(out_q + (rowbase + rr) * D_DIM + c4));
    v4f dd = e4 - x4;
    sq += dd.x * dd.x + dd.y * dd.y + dd.z * dd.z + dd.w * dd.w;
  }

  float cm = 0.f;
  if (tid < 64) {
    const int ind = s_am[tid];
    out_ind[rowbase + tid] = (float)ind;                 // indices as float output
    atomicAdd(&hist[ind], 1.f);
    cm = s_mn[tid] + s_xnorm[tid];                       // true dist at argmin
  }

  __syncthreads();
  s_redf[tid] = sq;
  __syncthreads();
  for (int off = 64; off > 0; off >>= 1) {
    if (tid < off) s_redf[tid] += s_redf[tid + off];
    __syncthreads();
  }
  if (tid == 0) atomicAdd(&accums[0], s_redf[0]);
  __syncthreads();
  s_redf[tid] = cm;
  __syncthreads();
  for (int off = 64; off > 0; off >>= 1) {
    if (tid < off) s_redf[tid] += s_redf[tid + off];
    __syncthreads();
  }
  if (tid == 0) atomicAdd(&accums[1], s_redf[0]);
}

// ---------------------------------------------------------------------------
// Finalize: entropy over histogram, scale scalar losses
// ---------------------------------------------------------------------------
__global__ __launch_bounds__(512)
void vq_finalize(const float* __restrict__ hist, const float* __restrict__ accums,
                 float* __restrict__ out_loss, float* __restrict__ out_ent,
                 float* __restrict__ out_cm, int Btot)
{
  __shared__ float red[512];
  const int t = threadIdx.x;
  const float invB = 1.f / (float)Btot;
  const float p = hist[t] * invB;
  red[t] = -p * logf(p + 1e-10f);
  __syncthreads();
  for (int off = 256; off > 0; off >>= 1) {
    if (t < off) red[t] += red[t + off];
    __syncthreads();
  }
  if (t == 0) {
    *out_ent  = red[0];
    *out_loss = accums[0] * (1.f + BETA_C) / ((float)Btot * (float)D_DIM);
    *out_cm   = accums[1] * invB;
  }
}

// ---------------------------------------------------------------------------
extern "C" void kernel_launch(void* const* d_in, const int* in_sizes, int n_in,
                              void* d_out, int out_size, void* d_ws, size_t ws_size,
                              hipStream_t stream)
{
  const float* lat = (const float*)d_in[0];
  const float* emb = (const float*)d_in[1];
  const int Btot = in_sizes[0] / D_DIM;   // 131072

  float* ws     = (float*)d_ws;
  float* enorm  = ws;                     // [512]
  float* hist   = ws + 512;               // [512]
  float* accums = ws + 1024;              // [2]: sum_sq_err, sum_min_dist

  float* out = (float*)d_out;
  const size_t loss_off = (size_t)Btot * D_DIM;
  const size_t ent_off  = loss_off + 1;
  const size_t ind_off  = ent_off + 1;
  const size_t soft_off = ind_off + (size_t)Btot;
  const size_t cm_off   = soft_off + (size_t)Btot * K_DIM;

  vq_init<<<1, 512, 0, stream>>>(emb, enorm, hist, accums);
  vq_main<<<Btot / M_ROWS, 128, 0, stream>>>(lat, emb, enorm, hist, accums,
                                             out /*quantized*/, out + ind_off,
                                             out + soft_off);
  vq_finalize<<<1, 512, 0, stream>>>(hist, accums, out + loss_off, out + ent_off,
                                     out + cm_off, Btot);
}